// ApplyRotary_27814208209209
// MI455X (gfx1250) — compile-verified
//
#include <hip/hip_runtime.h>

// RoPE forward, fp32, T=32768 H=32 D=128, rope_dim=64.
// Memory-bound: ~1 GiB traffic -> ~46us floor @ 23.3 TB/s. No WMMA applicable
// (no matrix contraction). CDNA5 paths: global_load_async_to_lds_b128
// (ASYNCcnt DMA) for sin/cos staging, s_wait_asynccnt, non-temporal b128
// loads/stores for the read-once/write-once stream.
//
// Scheduling: all streaming vmem (pair loads + passthrough copy) is issued
// BEFORE the LDS barrier so it overlaps the scalar cu_seqlen scan and the
// async staging latency; only the rope FMAs wait on LDS.

typedef float v4f __attribute__((ext_vector_type(4)));
typedef int   v4i __attribute__((ext_vector_type(4)));

// Exact parameter types for __builtin_amdgcn_global_load_async_to_lds_b128
// (arg0: non-const int4* in AS1; arg1: int4* in AS3).
typedef __attribute__((address_space(1))) v4i g_v4i;
typedef __attribute__((address_space(3))) v4i l_v4i;

constexpr int kH    = 32;
constexpr int kD    = 128;
constexpr int kRope = 64;
constexpr int kHD   = kH * kD; // 4096 floats per token

#if defined(__has_builtin)
#if __has_builtin(__builtin_amdgcn_global_load_async_to_lds_b128)
#define HAVE_ASYNC_LDS 1
#endif
#endif

__device__ __forceinline__ v4f nt_load4(const float* p) {
  return __builtin_nontemporal_load((const v4f*)p);
}
__device__ __forceinline__ void nt_store4(float* p, v4f v) {
  __builtin_nontemporal_store(v, (v4f*)p);
}

__global__ __launch_bounds__(256)
void ApplyRotary_27814208209209_kernel(const float* __restrict__ x,
                                       const float* __restrict__ sinc,
                                       const float* __restrict__ cosc,
                                       const int*   __restrict__ cu,
                                       float*       __restrict__ out,
                                       int n_seg)
{
  __shared__ __align__(16) float s_sin[kRope];
  __shared__ __align__(16) float s_cos[kRope];

  const int t   = blockIdx.x;     // one token per workgroup
  const int tid = threadIdx.x;
  const size_t tok_base = (size_t)t * kHD;

  // ---- 1) Early-issue all streaming loads (independent of sin/cos). ----
  // Pair item: thread owns (h, d0), d0 in [0,32); loads x[d0..+3] and its
  // rotation partner x[d0+32..+35] exactly once.
  const int h  = tid >> 3;        // 32 heads * 8 float4s = 256 pair items
  const int q  = tid & 7;
  const int d0 = q * 4;
  const size_t lo = tok_base + (size_t)h * kD + d0;

  v4f xlo = nt_load4(x + lo);        // x1 = x[d],    d in [0,32)
  v4f xhi = nt_load4(x + lo + 32);   // x2 = x[d+32]

  // Passthrough loads (d in [64,128)): 512 float4s per token, 2 per thread.
  size_t pidx[2];
  v4f    pval[2];
  #pragma unroll
  for (int k = 0; k < 2; ++k) {
    const int pt = tid + k * 256;         // [0,512)
    const int ph = pt >> 4;               // 16 float4s per head
    const int pr = pt & 15;
    pidx[k] = tok_base + (size_t)ph * kD + kRope + (size_t)pr * 4;
    pval[k] = nt_load4(x + pidx[k]);
  }

  // ---- 2) Uniform segment scan: base = largest cu[i] <= t (cu[0]==0). ----
  // t is workgroup-uniform -> scalarized to s_loads.
  int base_tok = 0;
  #pragma unroll 1
  for (int i = 1; i <= n_seg; ++i) {
    int c = cu[i];
    if (t >= c) base_tok = c;
  }
  const int off = t - base_tok;   // position within the sequence

  // ---- 3) Kick off async DMA of this token's sin/cos rows into LDS. ----
  // Wave 0 only: lanes 0..15 -> sin row, lanes 16..31 -> cos row, b128 each.
  if (tid < 32) {
    const int sq = tid & 15;
    const float* gsrc = (tid < 16) ? (sinc + (size_t)off * kRope)
                                   : (cosc + (size_t)off * kRope);
    float* ldst = (tid < 16) ? s_sin : s_cos;
#if defined(HAVE_ASYNC_LDS)
    __builtin_amdgcn_global_load_async_to_lds_b128(
        (g_v4i*)(gsrc + sq * 4),
        (l_v4i*)(ldst + sq * 4),
        /*offset=*/0, /*cpol=*/0);
#else
    *(v4f*)(ldst + sq * 4) = *(const v4f*)(gsrc + sq * 4);
#endif
  }

  // ---- 4) Passthrough stores overlap the in-flight DMA. ----
  #pragma unroll
  for (int k = 0; k < 2; ++k) {
    nt_store4(out + pidx[k], pval[k]);
  }

#if defined(HAVE_ASYNC_LDS)
  if (tid < 32) {                 // whole-wave (wave32) uniform branch
    asm volatile("s_wait_asynccnt 0" ::: "memory");
  }
#endif
  __syncthreads();

  // ---- 5) Rope math from LDS + final stores. ----
  {
    v4f slo = *(const v4f*)(s_sin + d0);
    v4f shi = *(const v4f*)(s_sin + 32 + d0);
    v4f clo = *(const v4f*)(s_cos + d0);
    v4f chi = *(const v4f*)(s_cos + 32 + d0);

    // out[d]    = x1*cos[d]    - x2*sin[d]       (rot = -x2)
    // out[d+32] = x2*cos[d+32] + x1*sin[d+32]    (rot = +x1)
    v4f olo = xlo * clo - xhi * slo;
    v4f ohi = xhi * chi + xlo * shi;

    nt_store4(out + lo,      olo);
    nt_store4(out + lo + 32, ohi);
  }
}

extern "C" void kernel_launch(void* const* d_in, const int* in_sizes, int n_in,
                              void* d_out, int out_size, void* d_ws, size_t ws_size,
                              hipStream_t stream) {
  const float* x    = (const float*)d_in[0];   // (T, H, D) fp32
  const float* sinc = (const float*)d_in[1];   // (MAX_SEQLEN, 64) fp32
  const float* cosc = (const float*)d_in[2];   // (MAX_SEQLEN, 64) fp32
  const int*   cu   = (const int*)d_in[3];     // (n_seg+1,) int32
  float*       out  = (float*)d_out;

  const int T     = in_sizes[0] / kHD;
  const int n_seg = in_sizes[3] - 1;

  dim3 grid(T), block(256);
  ApplyRotary_27814208209209_kernel<<<grid, block, 0, stream>>>(
      x, sinc, cosc, cu, out, n_seg);
}